// GINConv_31138512896562
// MI455X (gfx1250) — compile-verified
//
#include <hip/hip_runtime.h>
#include <hip/hip_bf16.h>

typedef __attribute__((ext_vector_type(2))) float v2f;
typedef __attribute__((ext_vector_type(8))) float v8f;

#define DIMF 128
#define LDA 132  // padded LDS row stride (floats): banks = (4m+kk)%64 -> conflict-free b64 reads

// ---------------- zero workspace ----------------
__global__ __launch_bounds__(256) void gin_zero_kernel(float* __restrict__ agg,
                                                       float* __restrict__ stats,
                                                       long n_agg) {
    long i = (long)blockIdx.x * blockDim.x + threadIdx.x;
    long stride = (long)gridDim.x * blockDim.x;
    for (long j = i; j < n_agg; j += stride) agg[j] = 0.0f;
    if (i < 2 * DIMF) stats[i] = 0.0f;
}

// ---------------- edge scatter-add (GIN aggregation) ----------------
__global__ __launch_bounds__(256) void gin_scatter_kernel(const float* __restrict__ x,
                                                          const long long* __restrict__ ei,
                                                          float* __restrict__ agg, int E) {
    long idx = (long)blockIdx.x * 256 + threadIdx.x;
    int e = (int)(idx >> 5);
    if (e >= E) return;
    int q = (int)(idx & 31);           // one wave per edge, 4 floats per lane
    long s = (long)ei[e];              // src row (wave-uniform; single cacheline req)
    long d = (long)ei[(long)E + e];    // dst row
    const float4 v = *(const float4*)(x + s * DIMF + q * 4);
    float* a = agg + d * DIMF + q * 4;
    atomicAdd(a + 0, v.x);
    atomicAdd(a + 1, v.y);
    atomicAdd(a + 2, v.z);
    atomicAdd(a + 3, v.w);
}

// ---------------- GEMM1: h1 = ((1+eps)x + agg) @ W1^T + b1, plus BN stats ----------------
__global__ __launch_bounds__(256) void gin_gemm1_kernel(const float* __restrict__ x,
                                                        const float* __restrict__ agg,
                                                        const float* __restrict__ epsp,
                                                        const float* __restrict__ W1,
                                                        const float* __restrict__ b1,
                                                        float* __restrict__ h1,
                                                        float* __restrict__ stats,
                                                        int nrows) {
    __shared__ float Alds[16 * LDA];
    __shared__ float csum[DIMF];
    __shared__ float csum2[DIMF];
    const int tid = threadIdx.x;
    const int row0 = blockIdx.x * 16;
    const bool full = (row0 + 16 <= nrows);   // block-uniform
    const float onePlusEps = 1.0f + epsp[0];

    if (tid < DIMF) csum[tid] = 0.0f; else csum2[tid - DIMF] = 0.0f;

    // stage A tile: A[m][k] = (1+eps)*x + agg
    if (full) {
        for (int i = tid; i < 16 * DIMF; i += 256) {
            int m = i >> 7, kk = i & (DIMF - 1);
            long r = row0 + m;
            Alds[m * LDA + kk] = onePlusEps * x[r * DIMF + kk] + agg[r * DIMF + kk];
        }
    } else {
        for (int i = tid; i < 16 * DIMF; i += 256) {
            int m = i >> 7, kk = i & (DIMF - 1);
            int r = row0 + m;
            float v = 0.0f;
            if (r < nrows) v = onePlusEps * x[(long)r * DIMF + kk] + agg[(long)r * DIMF + kk];
            Alds[m * LDA + kk] = v;
        }
    }
    __syncthreads();

    const int lane = tid & 31;
    const int wave = tid >> 5;
    const int hi = lane >> 4;      // lane group: K offset 0 or 2
    const int l16 = lane & 15;
    const int n = wave * 16 + l16; // output column

    v8f c = {};
    const float* Arow = Alds + l16 * LDA + 2 * hi;    // A[m][k+2hi .. +1]
    const float* Brow = W1 + (long)n * DIMF + 2 * hi; // B[k][n] = W1[n][k]
#pragma unroll 8
    for (int k = 0; k < DIMF; k += 4) {
        v2f a = *(const v2f*)(Arow + k);
        v2f b = *(const v2f*)(Brow + k);
        c = __builtin_amdgcn_wmma_f32_16x16x4_f32(false, a, false, b, (short)0, c,
                                                  false, false);
    }

    const float bias = b1[n];
    float s = 0.0f, s2 = 0.0f;
    if (full) {
        long base = (long)(row0 + 8 * hi) * DIMF + n;
#pragma unroll
        for (int r = 0; r < 8; ++r) {
            float h = c[r] + bias;
            h1[base + (long)r * DIMF] = h;
            s += h;
            s2 = fmaf(h, h, s2);
        }
    } else {
#pragma unroll
        for (int r = 0; r < 8; ++r) {
            int m = row0 + r + 8 * hi;
            if (m < nrows) {
                float h = c[r] + bias;
                h1[(long)m * DIMF + n] = h;
                s += h;
                s2 = fmaf(h, h, s2);
            }
        }
    }
    atomicAdd(&csum[n], s);   // LDS ds_add_f32; lanes l and l+16 share n
    atomicAdd(&csum2[n], s2);
    __syncthreads();
    if (tid < DIMF) atomicAdd(&stats[tid], csum[tid]);
    else            atomicAdd(&stats[tid], csum2[tid - DIMF]);
}

// ---------------- BN finalize: scale/shift per feature ----------------
__global__ __launch_bounds__(128) void gin_bn_kernel(const float* __restrict__ stats,
                                                     const float* __restrict__ gamma,
                                                     const float* __restrict__ beta,
                                                     float* __restrict__ bnp, float invN) {
    int t = threadIdx.x;
    float mean = stats[t] * invN;
    float ex2 = stats[DIMF + t] * invN;
    float var = ex2 - mean * mean;       // biased variance
    float sc = gamma[t] * rsqrtf(var + 1e-5f);
    bnp[t] = sc;
    bnp[DIMF + t] = beta[t] - mean * sc;
}

// ---------------- GEMM2: out = x + relu(relu(BN(h1)) @ W2^T + b2) ----------------
__global__ __launch_bounds__(256) void gin_gemm2_kernel(const float* __restrict__ h1,
                                                        const float* __restrict__ bnp,
                                                        const float* __restrict__ W2,
                                                        const float* __restrict__ b2,
                                                        const float* __restrict__ x,
                                                        float* __restrict__ out,
                                                        int nrows) {
    __shared__ float Alds[16 * LDA];
    const int tid = threadIdx.x;
    const int row0 = blockIdx.x * 16;
    const bool full = (row0 + 16 <= nrows);   // block-uniform

    // stage A tile: A[m][k] = relu(h1*scale + shift)
    if (full) {
        for (int i = tid; i < 16 * DIMF; i += 256) {
            int m = i >> 7, kk = i & (DIMF - 1);
            long r = row0 + m;
            float v = fmaf(h1[r * DIMF + kk], bnp[kk], bnp[DIMF + kk]);
            Alds[m * LDA + kk] = fmaxf(v, 0.0f);
        }
    } else {
        for (int i = tid; i < 16 * DIMF; i += 256) {
            int m = i >> 7, kk = i & (DIMF - 1);
            int r = row0 + m;
            float v = 0.0f;
            if (r < nrows)
                v = fmaxf(fmaf(h1[(long)r * DIMF + kk], bnp[kk], bnp[DIMF + kk]), 0.0f);
            Alds[m * LDA + kk] = v;
        }
    }
    __syncthreads();

    const int lane = tid & 31;
    const int wave = tid >> 5;
    const int hi = lane >> 4;
    const int l16 = lane & 15;
    const int n = wave * 16 + l16;

    v8f c = {};
    const float* Arow = Alds + l16 * LDA + 2 * hi;
    const float* Brow = W2 + (long)n * DIMF + 2 * hi;
#pragma unroll 8
    for (int k = 0; k < DIMF; k += 4) {
        v2f a = *(const v2f*)(Arow + k);
        v2f b = *(const v2f*)(Brow + k);
        c = __builtin_amdgcn_wmma_f32_16x16x4_f32(false, a, false, b, (short)0, c,
                                                  false, false);
    }

    const float bias = b2[n];
    if (full) {
        long base = (long)(row0 + 8 * hi) * DIMF + n;
#pragma unroll
        for (int r = 0; r < 8; ++r) {
            float h = fmaxf(c[r] + bias, 0.0f);
            out[base + (long)r * DIMF] = x[base + (long)r * DIMF] + h;
        }
    } else {
#pragma unroll
        for (int r = 0; r < 8; ++r) {
            int m = row0 + r + 8 * hi;
            if (m < nrows) {
                float h = fmaxf(c[r] + bias, 0.0f);
                out[(long)m * DIMF + n] = x[(long)m * DIMF + n] + h;
            }
        }
    }
}

extern "C" void kernel_launch(void* const* d_in, const int* in_sizes, int n_in,
                              void* d_out, int out_size, void* d_ws, size_t ws_size,
                              hipStream_t stream) {
    const float* x         = (const float*)d_in[0];
    const long long* ei    = (const long long*)d_in[1];  // int64 [2,E]
    const float* eps       = (const float*)d_in[2];
    const float* W1        = (const float*)d_in[3];
    const float* b1        = (const float*)d_in[4];
    const float* gamma     = (const float*)d_in[5];
    const float* beta      = (const float*)d_in[6];
    const float* W2        = (const float*)d_in[7];
    const float* b2        = (const float*)d_in[8];
    float* out             = (float*)d_out;

    const int nrows = in_sizes[0] / DIMF;
    const int E     = in_sizes[1] / 2;

    // workspace layout (floats): agg[N*D] | h1[N*D] | stats[256] | bnp[256]
    float* agg   = (float*)d_ws;
    float* h1    = agg + (long)nrows * DIMF;
    float* stats = h1 + (long)nrows * DIMF;
    float* bnp   = stats + 2 * DIMF;

    gin_zero_kernel<<<4096, 256, 0, stream>>>(agg, stats, (long)nrows * DIMF);

    long nscatter = (long)E * 32;
    gin_scatter_kernel<<<(int)((nscatter + 255) / 256), 256, 0, stream>>>(x, ei, agg, E);

    int ntiles = (nrows + 15) / 16;
    gin_gemm1_kernel<<<ntiles, 256, 0, stream>>>(x, agg, eps, W1, b1, h1, stats, nrows);
    gin_bn_kernel<<<1, 128, 0, stream>>>(stats, gamma, beta, bnp, 1.0f / (float)nrows);
    gin_gemm2_kernel<<<ntiles, 256, 0, stream>>>(h1, bnp, W2, b2, x, out, nrows);
}